// GraphConvLayer_41515153883727
// MI455X (gfx1250) — compile-verified
//
#include <hip/hip_runtime.h>
#include <hip/hip_bf16.h>

typedef __attribute__((ext_vector_type(2))) float v2f;
typedef __attribute__((ext_vector_type(8))) float v8f;

#define IN_CH   128
#define OUT_CH  128
#define LDS_PAD 4
#define A_STRIDE (IN_CH + LDS_PAD)   // 132: breaks stride-128 bank conflicts
#define W_STRIDE (OUT_CH + LDS_PAD)  // 132

// ---------------------------------------------------------------- zero init
__global__ void gcn_zero_f(float* __restrict__ p, int n) {
    int i = blockIdx.x * blockDim.x + threadIdx.x;
    if (i < n) p[i] = 0.0f;
}

// ---------------------------------------------------------------- degrees
__global__ void gcn_degree(const int* __restrict__ ei, float* __restrict__ deg, int E) {
    int e = blockIdx.x * blockDim.x + threadIdx.x;
    if (e < E) atomicAdd(&deg[ei[e]], 1.0f);   // row = ei[0:E]
}

// ---------------------------------------------------------------- deg -> deg^-1/2 (in place)
__global__ void gcn_dinv(float* __restrict__ deg, int N) {
    int i = blockIdx.x * blockDim.x + threadIdx.x;
    if (i < N) {
        float d = deg[i];
        d = (d == 0.0f) ? 1.0f : d;
        deg[i] = rsqrtf(d);
    }
}

// ---------------------------------------------------------------- xw = x @ W via V_WMMA_F32_16X16X4_F32
// One block = 256 threads = 8 wave32s. Block owns 16 rows; each wave owns a
// 16-wide column tile (8 waves * 16 = 128 = OUT_CH). K=128 -> 32 WMMA k-steps.
__global__ void __launch_bounds__(256, 2)
gcn_gemm_xw(const float* __restrict__ x, const float* __restrict__ w,
            float* __restrict__ xw, int N) {
    __shared__ float ldsW[IN_CH * W_STRIDE];  // weight, [k][n] padded
    __shared__ float ldsA[16 * A_STRIDE];     // x tile,  [m][k] padded

    const int tid     = threadIdx.x;
    const int rowBase = blockIdx.x * 16;

    // Stage weight (128x128 f32 = 64KB) via coalesced float4 loads.
    for (int i = tid; i < (IN_CH * OUT_CH) / 4; i += 256) {
        float4 v = ((const float4*)w)[i];
        int k = (i * 4) / OUT_CH;
        int n = (i * 4) % OUT_CH;
        float* dst = &ldsW[k * W_STRIDE + n];
        dst[0] = v.x; dst[1] = v.y; dst[2] = v.z; dst[3] = v.w;
    }
    // Stage A tile (16x128 f32 = 8KB), clamped for the (unused) tail case.
    for (int i = tid; i < (16 * IN_CH) / 4; i += 256) {
        int m = (i * 4) / IN_CH;
        int k = (i * 4) % IN_CH;
        int r = rowBase + m; if (r >= N) r = N - 1;
        float4 v = *(const float4*)(x + (size_t)r * IN_CH + k);
        float* dst = &ldsA[m * A_STRIDE + k];
        dst[0] = v.x; dst[1] = v.y; dst[2] = v.z; dst[3] = v.w;
    }
    __syncthreads();

    const int wave = tid >> 5;
    const int lane = tid & 31;
    const int lo   = lane & 15;   // M (for A) / N (for B)
    const int hi   = lane >> 4;   // selects K pair {k, k+1} vs {k+2, k+3}
    const int colBase = wave * 16;

    // ISA 16x4 f32 A layout: lane m in [0,15] holds K=k,k+1; lanes 16-31 hold K=k+2,k+3
    const float* ar = ldsA + lo * A_STRIDE + 2 * hi;
    const float* br = ldsW + (2 * hi) * W_STRIDE + colBase + lo;

    v8f c = {};
#pragma unroll
    for (int k = 0; k < IN_CH; k += 4) {
        v2f a, b;
        a.x = ar[k];
        a.y = ar[k + 1];
        b.x = br[k * W_STRIDE];
        b.y = br[k * W_STRIDE + W_STRIDE];
        c = __builtin_amdgcn_wmma_f32_16x16x4_f32(
                /*neg_a=*/false, a, /*neg_b=*/false, b,
                /*c_mod=*/(short)0, c, /*reuse_a=*/false, /*reuse_b=*/false);
    }

    // C/D layout: VGPR i -> row (rowBase + i + 8*hi), col (colBase + lo)
    const int rBase = rowBase + 8 * hi;
    float* orow = xw + (size_t)rBase * OUT_CH + colBase + lo;
#pragma unroll
    for (int i = 0; i < 8; ++i) {
        if (rBase + i < N) orow[(size_t)i * OUT_CH] = c[i];
    }
}

// ---------------------------------------------------------------- edge scatter
// One wave32 per edge: out[row] += dinv[row]*dinv[col] * xw[col]  (128 f32 = 1 float4/lane)
__global__ void __launch_bounds__(256)
gcn_scatter(const float4* __restrict__ xw, const float* __restrict__ dinv,
            const int* __restrict__ ei, float* __restrict__ out, int E) {
    int wave = blockIdx.x * 8 + (threadIdx.x >> 5);
    int lane = threadIdx.x & 31;
    if (wave >= E) return;

    int row = ei[wave];       // target
    int col = ei[E + wave];   // source
    float coef = dinv[row] * dinv[col];

    float4 v = xw[(size_t)col * (IN_CH / 4) + lane];  // 32 lanes * float4 = 128 ch
    float* o = out + (size_t)row * OUT_CH + lane * 4;
    atomicAdd(o + 0, v.x * coef);
    atomicAdd(o + 1, v.y * coef);
    atomicAdd(o + 2, v.z * coef);
    atomicAdd(o + 3, v.w * coef);
}

// ---------------------------------------------------------------- launcher
extern "C" void kernel_launch(void* const* d_in, const int* in_sizes, int n_in,
                              void* d_out, int out_size, void* d_ws, size_t ws_size,
                              hipStream_t stream) {
    const float* x  = (const float*)d_in[0];   // [N, 128]
    const float* w  = (const float*)d_in[1];   // [128, 128]
    const int*   ei = (const int*)d_in[2];     // [2, E] flat: rows then cols
    float*       out = (float*)d_out;          // [N, 128]

    const int N = in_sizes[0] / IN_CH;
    const int E = in_sizes[2] / 2;

    // Workspace layout: [deg/dinv : N floats][pad to 256B][xw : N*128 floats]
    float* deg = (float*)d_ws;
    size_t xw_off = (((size_t)N * sizeof(float)) + 255) & ~(size_t)255;
    float* xw = (float*)((char*)d_ws + xw_off);

    // 1) zero deg and the output accumulator
    gcn_zero_f<<<(N + 255) / 256, 256, 0, stream>>>(deg, N);
    gcn_zero_f<<<((N * OUT_CH) + 255) / 256, 256, 0, stream>>>(out, N * OUT_CH);

    // 2) xw = x @ W  (WMMA fp32)
    gcn_gemm_xw<<<(N + 15) / 16, 256, 0, stream>>>(x, w, xw, N);

    // 3) degrees, 4) dinv
    gcn_degree<<<(E + 255) / 256, 256, 0, stream>>>(ei, deg, E);
    gcn_dinv<<<(N + 255) / 256, 256, 0, stream>>>(deg, N);

    // 5) edge scatter with L2-resident atomics
    gcn_scatter<<<(E + 7) / 8, 256, 0, stream>>>((const float4*)xw, deg, ei, out, E);
}